// ODE2VAE_6262062317918
// MI455X (gfx1250) — compile-verified
//
#include <hip/hip_runtime.h>

typedef __attribute__((ext_vector_type(2))) float v2f;
typedef __attribute__((ext_vector_type(8))) float v8f;

#define DD   64      // state dim (2Q)
#define HH   50      // hidden
#define QQ   32      // q
#define SAMP 64      // samples per workgroup
#define STRD 65      // LDS row stride (floats) -> conflict-free banks
#define DTf  0.1f
#define NSTEP 16

// ---------- LDS layout (float offsets) ----------
#define L_W1  0                         // 64x64 padded W1, stride 65
#define L_W2  (64*STRD)                 // padded W2
#define L_W3  (2*64*STRD)               // padded W3
#define L_C   (3*64*STRD)               // C = W3 @ W1[:32,:], padded
#define L_G   (4*64*STRD)               // Gt[k][l] = C[l][k]*W2[k][l]
#define L_CB  (5*64*STRD)               // consts: b1,g1,be1,b2,g2,be2,b3,rc,rw2 (9x64)
#define L_X   (L_CB + 9*64)             // state [SAMP][STRD]
#define L_U1  (L_X  + SAMP*STRD)
#define L_E1  (L_U1 + SAMP*STRD)
#define L_U2  (L_E1 + SAMP*STRD)
#define L_E2  (L_U2 + SAMP*STRD)
#define L_HB  (L_E2 + SAMP*STRD)        // h1 then h2
#define L_SCR (L_HB + SAMP*STRD)        // GEMM output scratch
#define L_TMP (L_SCR+ SAMP*STRD)        // saved GEMM output (XC / YC)
#define L_LP  (L_TMP+ SAMP*STRD)        // logp [SAMP]
#define L_END (L_LP + SAMP)
#define LDS_BYTES (L_END * 4)

// ---------- ws layout (float offsets, stride 64) ----------
// W1p@0, W2p@4096, W3p@8192, Cp@12288, Gt@16384, consts@20480 (9*64)
#define WS_FLOATS (5*4096 + 9*64)

// ============================================================
// Prep kernel: pad weights, build C, Gt, row sums. One block.
// ============================================================
__global__ void __launch_bounds__(256) ode2vae_prep(
    const float* __restrict__ W1, const float* __restrict__ W2,
    const float* __restrict__ W3, const float* __restrict__ b1,
    const float* __restrict__ g1, const float* __restrict__ be1,
    const float* __restrict__ b2, const float* __restrict__ g2,
    const float* __restrict__ be2, const float* __restrict__ b3,
    float* __restrict__ ws)
{
    const int tid = threadIdx.x;
    for (int i = tid; i < 4096; i += 256) {           // W1p: (64 x 50)->64x64
        int r = i >> 6, c = i & 63;
        ws[i] = (c < HH) ? W1[r * HH + c] : 0.f;
    }
    for (int i = tid; i < 4096; i += 256) {           // W2p
        int r = i >> 6, c = i & 63;
        ws[4096 + i] = (r < HH && c < HH) ? W2[r * HH + c] : 0.f;
    }
    for (int i = tid; i < 4096; i += 256) {           // W3p: (50 x 32)
        int r = i >> 6, c = i & 63;
        ws[8192 + i] = (r < HH && c < QQ) ? W3[r * QQ + c] : 0.f;
    }
    for (int i = tid; i < 4096; i += 256) {           // C[l][k] = sum_q W3[l][q]*W1[q][k]
        int l = i >> 6, k = i & 63;
        float v = 0.f;
        if (l < HH && k < HH)
            for (int q = 0; q < QQ; ++q) v += W3[l * QQ + q] * W1[q * HH + k];
        ws[12288 + i] = v;
    }
    __syncthreads();
    for (int i = tid; i < 4096; i += 256) {           // Gt[k][l] = C[l][k]*W2[k][l]
        int k = i >> 6, l = i & 63;
        ws[16384 + i] = ws[12288 + l * 64 + k] * ws[4096 + k * 64 + l];
    }
    if (tid < 64) {
        int c = tid;
        const int CB = 20480;
        ws[CB +   0 + c] = (c < HH) ? b1[c]  : 0.f;
        ws[CB +  64 + c] = (c < HH) ? g1[c]  : 0.f;
        ws[CB + 128 + c] = (c < HH) ? be1[c] : 0.f;
        ws[CB + 192 + c] = (c < HH) ? b2[c]  : 0.f;
        ws[CB + 256 + c] = (c < HH) ? g2[c]  : 0.f;
        ws[CB + 320 + c] = (c < HH) ? be2[c] : 0.f;
        ws[CB + 384 + c] = (c < QQ) ? b3[c]  : 0.f;
        float rc = 0.f, rw = 0.f;
        if (c < HH) {
            for (int k = 0; k < HH; ++k) rc += ws[12288 + c * 64 + k]; // rowsum C
            for (int l = 0; l < HH; ++l) rw += W2[c * HH + l];         // rowsum W2
        }
        ws[CB + 448 + c] = rc;
        ws[CB + 512 + c] = rw;
    }
}

// ============================================================
// WMMA tile GEMM: D[64xSTRD] = A(64x64, LDS) @ B(64x64, LDS)
// wave w computes M-tile w. KT k-steps of 4, NT n-tiles of 16.
// TRB: read B transposed in-place. PROD: A = A0 .* Am elementwise.
// ============================================================
template <int KT, int NT, bool TRB, bool PROD>
__device__ __forceinline__ void gemm_tile(const float* __restrict__ A0,
                                          const float* __restrict__ Am,
                                          const float* __restrict__ B,
                                          float* __restrict__ D,
                                          int wave, int lane)
{
    const int mrow = wave * 16 + (lane & 15);
    const int ksel = (lane >> 4) << 1;   // lanes 0-15: +0 ; lanes 16-31: +2
    const int ncol = lane & 15;
    v2f afr[KT];
#pragma unroll
    for (int kk = 0; kk < KT; ++kk) {    // A 16x4 f32 layout: v0 K=k0, v1 K=k0+1
        const int k0 = kk * 4 + ksel;
        float ax = A0[mrow * STRD + k0];
        float ay = A0[mrow * STRD + k0 + 1];
        if (PROD) { ax *= Am[mrow * STRD + k0]; ay *= Am[mrow * STRD + k0 + 1]; }
        afr[kk].x = ax; afr[kk].y = ay;
    }
#pragma unroll
    for (int nt = 0; nt < NT; ++nt) {
        const int n = nt * 16 + ncol;
        v8f acc = {0.f, 0.f, 0.f, 0.f, 0.f, 0.f, 0.f, 0.f};
#pragma unroll
        for (int kk = 0; kk < KT; ++kk) {
            const int k0 = kk * 4 + ksel;
            v2f b;
            if (TRB) { b.x = B[n * STRD + k0];  b.y = B[n * STRD + k0 + 1]; }
            else     { b.x = B[k0 * STRD + n];  b.y = B[(k0 + 1) * STRD + n]; }
            acc = __builtin_amdgcn_wmma_f32_16x16x4_f32(
                false, afr[kk], false, b, (short)0, acc, false, false);
        }
        const int mb = wave * 16 + ((lane >> 4) << 3);  // C/D: v_j -> rows j / j+8
#pragma unroll
        for (int j = 0; j < 8; ++j) D[(mb + j) * STRD + n] = acc[j];
    }
}

// ============================================================
// Main kernel: 64 samples per block, 16 Euler steps in LDS.
// ============================================================
__global__ void __launch_bounds__(128, 1) ode2vae_main(
    const float* __restrict__ z0, const float* __restrict__ logp0,
    const float* __restrict__ ws, float* __restrict__ out, int N)
{
    extern __shared__ float lds[];
    const int tid  = threadIdx.x;
    const int wave = tid >> 5;
    const int lane = tid & 31;
    const long n0  = (long)blockIdx.x * SAMP;

    // load weights (restride 64 -> 65) + consts + state
    for (int i = tid; i < 5 * 4096; i += 128) {
        int mat = i >> 12, idx = i & 4095, r = idx >> 6, c = idx & 63;
        lds[mat * (64 * STRD) + r * STRD + c] = ws[i];
    }
    for (int i = tid; i < 9 * 64; i += 128) lds[L_CB + i] = ws[20480 + i];
    for (int i = tid; i < SAMP * DD; i += 128) {
        int s = i >> 6, j = i & 63;
        lds[L_X + s * STRD + j] = z0[(n0 + s) * DD + j];
    }
    if (tid < SAMP) lds[L_LP + tid] = logp0[n0 + tid];
    __syncthreads();

    float* sX  = lds + L_X;   float* sU1 = lds + L_U1;  float* sE1 = lds + L_E1;
    float* sU2 = lds + L_U2;  float* sE2 = lds + L_E2;  float* sHB = lds + L_HB;
    float* sSC = lds + L_SCR; float* sTM = lds + L_TMP;
    const float* cb1 = lds + L_CB;        const float* cg1 = lds + L_CB + 64;
    const float* ce1 = lds + L_CB + 128;  const float* cb2 = lds + L_CB + 192;
    const float* cg2 = lds + L_CB + 256;  const float* ce2 = lds + L_CB + 320;
    const float* cb3 = lds + L_CB + 384;  const float* crc = lds + L_CB + 448;
    const float* crw = lds + L_CB + 512;

    for (int step = 0; step < NSTEP; ++step) {
        float S1=0,S2=0,S3=0,S4=0,S5=0,S6=0,S7=0,S8=0;
        float TA=0,TB1=0,TB2=0,TC1=0,TC2=0;

        // ---- layer 1: A1 = X @ W1p ----
        gemm_tile<16,4,false,false>(sX, nullptr, lds + L_W1, sSC, wave, lane);
        __syncthreads();
        if (tid < SAMP) {  // layernorm + celu; e1 = r*g*celu'(y); S1,S3
            float* row = sSC + tid * STRD;
            float m = 0.f;
            for (int h = 0; h < HH; ++h) m += row[h] + cb1[h];
            m *= (1.f / HH);
            float var = 0.f;
            for (int h = 0; h < HH; ++h) { float a = row[h] + cb1[h] - m; var += a * a; }
            const float r = rsqrtf(var * (1.f / HH) + 1e-5f);
            for (int h = 0; h < DD; ++h) {
                if (h < HH) {
                    float u = (row[h] + cb1[h] - m) * r;
                    float y = u * cg1[h] + ce1[h];
                    float ex = __expf(y);
                    float phi = (y > 0.f) ? 1.f : ex;
                    float e = r * cg1[h] * phi;
                    sU1[tid * STRD + h] = u;
                    sE1[tid * STRD + h] = e;
                    sHB[tid * STRD + h] = (y > 0.f) ? y : (ex - 1.f);
                    S1 += e * crw[h];
                    S3 += e * u * crw[h];
                } else {
                    sU1[tid * STRD + h] = 0.f; sE1[tid * STRD + h] = 0.f;
                    sHB[tid * STRD + h] = 0.f;
                }
            }
        }
        __syncthreads();

        // ---- layer 2: A2 = H1 @ W2p ----
        gemm_tile<13,4,false,false>(sHB, nullptr, lds + L_W2, sSC, wave, lane);
        __syncthreads();
        if (tid < SAMP) {
            float* row = sSC + tid * STRD;
            float m = 0.f;
            for (int h = 0; h < HH; ++h) m += row[h] + cb2[h];
            m *= (1.f / HH);
            float var = 0.f;
            for (int h = 0; h < HH; ++h) { float a = row[h] + cb2[h] - m; var += a * a; }
            const float r = rsqrtf(var * (1.f / HH) + 1e-5f);
            for (int h = 0; h < DD; ++h) {
                if (h < HH) {
                    float u = (row[h] + cb2[h] - m) * r;
                    float y = u * cg2[h] + ce2[h];
                    float ex = __expf(y);
                    float phi = (y > 0.f) ? 1.f : ex;
                    float e = r * cg2[h] * phi;
                    sU2[tid * STRD + h] = u;
                    sE2[tid * STRD + h] = e;
                    sHB[tid * STRD + h] = (y > 0.f) ? y : (ex - 1.f);  // h2 over h1
                    S2 += e * crc[h];
                    S6 += e * u * crc[h];
                } else {
                    sU2[tid * STRD + h] = 0.f; sE2[tid * STRD + h] = 0.f;
                    sHB[tid * STRD + h] = 0.f;
                }
            }
        }
        __syncthreads();

        // ---- trace GEMMs (closed-form log-det trace) ----
        // XC = U1 @ C^T  -> keep in TMP; S4 = e2.XC ; S8 = e2*u2.XC
        gemm_tile<13,4,true,false>(sU1, nullptr, lds + L_C, sSC, wave, lane);
        __syncthreads();
        if (tid < SAMP) {
            for (int h = 0; h < HH; ++h) {
                float xc = sSC[tid * STRD + h];
                sTM[tid * STRD + h] = xc;
                float e2 = sE2[tid * STRD + h];
                S4 += e2 * xc;
                S8 += e2 * sU2[tid * STRD + h] * xc;
            }
        }
        __syncthreads();
        // YB = (E1.*U1) @ W2 ;  TC2 = sum e2 * XC * YB
        gemm_tile<13,4,false,true>(sE1, sU1, lds + L_W2, sSC, wave, lane);
        __syncthreads();
        if (tid < SAMP)
            for (int h = 0; h < HH; ++h)
                TC2 += sE2[tid * STRD + h] * sTM[tid * STRD + h] * sSC[tid * STRD + h];
        __syncthreads();
        // YC = U2 @ W2^T -> keep in TMP; S5 = e1.YC ; S7 = e1*u1.YC
        gemm_tile<13,4,true,false>(sU2, nullptr, lds + L_W2, sSC, wave, lane);
        __syncthreads();
        if (tid < SAMP) {
            for (int h = 0; h < HH; ++h) {
                float yc = sSC[tid * STRD + h];
                sTM[tid * STRD + h] = yc;
                float e1 = sE1[tid * STRD + h];
                S5 += e1 * yc;
                S7 += e1 * sU1[tid * STRD + h] * yc;
            }
        }
        __syncthreads();
        // XB = (E2.*U2) @ C ;  TB2 = sum e1 * YC * XB
        gemm_tile<13,4,false,true>(sE2, sU2, lds + L_C, sSC, wave, lane);
        __syncthreads();
        if (tid < SAMP)
            for (int h = 0; h < HH; ++h)
                TB2 += sE1[tid * STRD + h] * sTM[tid * STRD + h] * sSC[tid * STRD + h];
        __syncthreads();
        // XA = E2 @ C ;  TB1 = sum e1 * rw2 * XA
        gemm_tile<13,4,false,false>(sE2, nullptr, lds + L_C, sSC, wave, lane);
        __syncthreads();
        if (tid < SAMP)
            for (int h = 0; h < HH; ++h)
                TB1 += sE1[tid * STRD + h] * crw[h] * sSC[tid * STRD + h];
        __syncthreads();
        // YA = E1 @ W2 ;  TC1 = sum e2 * rc * YA
        gemm_tile<13,4,false,false>(sE1, nullptr, lds + L_W2, sSC, wave, lane);
        __syncthreads();
        if (tid < SAMP)
            for (int h = 0; h < HH; ++h)
                TC1 += sE2[tid * STRD + h] * crc[h] * sSC[tid * STRD + h];
        __syncthreads();
        // V1 = E1 @ Gt ;  TA = sum e2 * V1
        gemm_tile<13,4,false,false>(sE1, nullptr, lds + L_G, sSC, wave, lane);
        __syncthreads();
        if (tid < SAMP)
            for (int h = 0; h < HH; ++h)
                TA += sE2[tid * STRD + h] * sSC[tid * STRD + h];
        __syncthreads();

        // ---- output layer: DV = H2 @ W3p (cols < 32) ----
        gemm_tile<13,2,false,false>(sHB, nullptr, lds + L_W3, sSC, wave, lane);
        __syncthreads();

        if (tid < SAMP) {
            const float iH = 1.f / (float)HH;
            float tr = TA - (TB1 + TB2) * iH - (TC1 + TC2) * iH
                     + (S1 * S2 + S3 * S4 + S5 * S6 + S7 * S8) * iH * iH;
            float lp = lds[L_LP + tid] - DTf * tr;
            lds[L_LP + tid] = lp;
            const size_t ob = ((size_t)step * (size_t)N + (size_t)(n0 + tid)) * 65;
            for (int j = 0; j < QQ; ++j) {
                float dv = sSC[tid * STRD + j] + cb3[j];
                float v  = sX[tid * STRD + j];
                float s  = sX[tid * STRD + QQ + j];
                float vn = v + DTf * dv;
                float sn = s + DTf * v;
                sX[tid * STRD + j]      = vn;
                sX[tid * STRD + QQ + j] = sn;
                out[ob + j]      = vn;
                out[ob + QQ + j] = sn;
            }
            out[ob + 64] = lp;
        }
        __syncthreads();
    }
}

extern "C" void kernel_launch(void* const* d_in, const int* in_sizes, int n_in,
                              void* d_out, int out_size, void* d_ws, size_t ws_size,
                              hipStream_t stream)
{
    const float* z0  = (const float*)d_in[0];
    const float* lp0 = (const float*)d_in[1];
    const float* W1  = (const float*)d_in[2];
    const float* b1  = (const float*)d_in[3];
    const float* g1  = (const float*)d_in[4];
    const float* be1 = (const float*)d_in[5];
    const float* W2  = (const float*)d_in[6];
    const float* b2  = (const float*)d_in[7];
    const float* g2  = (const float*)d_in[8];
    const float* be2 = (const float*)d_in[9];
    const float* W3  = (const float*)d_in[10];
    const float* b3  = (const float*)d_in[11];
    float* ws  = (float*)d_ws;
    float* out = (float*)d_out;
    const int N = in_sizes[0] / DD;          // 131072
    (void)n_in; (void)out_size; (void)ws_size;

    ode2vae_prep<<<1, 256, 0, stream>>>(W1, W2, W3, b1, g1, be1, b2, g2, be2, b3, ws);

    hipFuncSetAttribute(reinterpret_cast<const void*>(ode2vae_main),
                        hipFuncAttributeMaxDynamicSharedMemorySize, LDS_BYTES);
    ode2vae_main<<<N / SAMP, 128, LDS_BYTES, stream>>>(z0, lp0, ws, out, N);
}